// Model_70446053589461
// MI455X (gfx1250) — compile-verified
//
#include <hip/hip_runtime.h>
#include <cmath>

// RNN: h_t = tanh(x_t @ W_ih^T + b_ih + b_hh + h_{t-1} @ W_hh^T)
// S=2048, B=64, I=H=512.  FP32 path via V_WMMA_F32_16X16X4_F32 (wave32),
// LDS staging via GLOBAL_LOAD_ASYNC_TO_LDS_B128 (ASYNCcnt).

#define S_LEN 2048
#define B_DIM 64
#define I_DIM 512
#define H_DIM 512
#define NWG   32          // workgroups in the persistent scan kernel
#define WPAD  516         // 512 + 4 pad floats -> LDS row stride hits all 64 banks
#define KB    32          // K-chunk staged per LDS round in xp_gemm
#define LSTR  36          // LDS row stride (floats): 144B = 16B aligned, 36n mod 64 distinct

typedef float v2f __attribute__((ext_vector_type(2)));
typedef float v8f __attribute__((ext_vector_type(8)));

__device__ __forceinline__ v8f wmma_f32_k4(v2f a, v2f b, v8f c) {
    // D = A(16x4) * B(4x16) + C, full fp32.
    return __builtin_amdgcn_wmma_f32_16x16x4_f32(false, a, false, b,
                                                 (short)0, c, false, false);
}

// Per-lane 16B global -> LDS async copy (GV mode), tracked by ASYNCcnt.
__device__ __forceinline__ void async_g2l_b128(unsigned lds_off, const void* gptr) {
    asm volatile("global_load_async_to_lds_b128 %0, %1, off"
                 :: "v"(lds_off), "v"((unsigned long long)gptr) : "memory");
}
__device__ __forceinline__ void wait_async0() {
    asm volatile("s_wait_asynccnt 0" ::: "memory");
}

// ---------------------------------------------------------------------------
// Kernel 0: zero the h0 exchange buffer and the grid-barrier counter (d_ws is
// poisoned with 0xAA before timing; must re-init every call for determinism).
// ---------------------------------------------------------------------------
__global__ void init_ws(float* __restrict__ hbuf0, unsigned* __restrict__ counter) {
    int i = blockIdx.x * blockDim.x + threadIdx.x;
    if (i == 0) *counter = 0u;
    if (i < B_DIM * H_DIM) hbuf0[i] = 0.0f;
}

// ---------------------------------------------------------------------------
// Kernel 1: xp[m,n] = sum_k x[m,k]*W_ih[n,k] + b_ih[n] + b_hh[n]
// M = S*B = 131072, N = 512, K = 512.  Block = 256 threads computes a 128x128
// output tile; A(128xKB) and B(128xKB) chunks are async-staged into LDS and
// shared by all 8 waves (wave = 32Mx64N = 2x4 accumulator tiles).  W_ih is
// read once per 128-row block (~1 GB total instead of ~8.6 TB from L2).
// ---------------------------------------------------------------------------
__global__ __launch_bounds__(256) void xp_gemm(const float* __restrict__ x,
                                               const float* __restrict__ Wih,
                                               const float* __restrict__ bih,
                                               const float* __restrict__ bhh,
                                               float* __restrict__ xp) {
    __shared__ float As[128 * LSTR];   // x chunk, row-major [m][k], padded
    __shared__ float Bs[128 * LSTR];   // W_ih chunk, row-major [n][k], padded

    const int tid  = threadIdx.x;
    const int wave = tid >> 5;
    const int lane = tid & 31;
    const int lr   = lane & 15;
    const int hi   = lane >> 4;
    const int wm   = wave & 3;     // 4 row groups of 32
    const int wn   = wave >> 2;    // 2 col groups of 64
    const int M0   = blockIdx.x * 128;
    const int N0   = blockIdx.y * 128;

    v8f c[2][4];
#pragma unroll
    for (int i = 0; i < 2; ++i)
#pragma unroll
        for (int j = 0; j < 4; ++j) c[i][j] = (v8f){0,0,0,0,0,0,0,0};

    const float* abase = As + (size_t)(32 * wm + lr) * LSTR + 2 * hi;
    const float* bbase = Bs + (size_t)(64 * wn + lr) * LSTR + 2 * hi;

    for (int k0 = 0; k0 < I_DIM; k0 += KB) {
        // Stage A and B chunks: 128 rows x 32 floats each = 1024 x b128
        // transfers per matrix; 4 per thread per matrix.
#pragma unroll
        for (int j = 0; j < 4; ++j) {
            const int i   = tid + 256 * j;     // 0..1023
            const int row = i >> 3;            // 0..127
            const int cg  = (i & 7) * 4;       // float offset in row, 16B steps
            async_g2l_b128((unsigned)(size_t)(As + row * LSTR + cg),
                           x + (size_t)(M0 + row) * I_DIM + k0 + cg);
            async_g2l_b128((unsigned)(size_t)(Bs + row * LSTR + cg),
                           Wih + (size_t)(N0 + row) * I_DIM + k0 + cg);
        }
        wait_async0();
        __syncthreads();

#pragma unroll
        for (int kl = 0; kl < KB; kl += 4) {
            v2f a0 = *(const v2f*)(abase + kl);
            v2f a1 = *(const v2f*)(abase + 16 * LSTR + kl);
#pragma unroll
            for (int j = 0; j < 4; ++j) {
                v2f b = *(const v2f*)(bbase + j * 16 * LSTR + kl);
                c[0][j] = wmma_f32_k4(a0, b, c[0][j]);
                c[1][j] = wmma_f32_k4(a1, b, c[1][j]);
            }
        }
        __syncthreads();   // protect LDS before next chunk overwrites
    }

#pragma unroll
    for (int j = 0; j < 4; ++j) {
        const int n = N0 + 64 * wn + 16 * j + lr;
        const float bias = bih[n] + bhh[n];
#pragma unroll
        for (int i = 0; i < 2; ++i) {
            const int mb = M0 + 32 * wm + 16 * i;
#pragma unroll
            for (int r = 0; r < 8; ++r)
                xp[(size_t)(mb + r + 8 * hi) * H_DIM + n] = c[i][j][r] + bias;
        }
    }
}

// ---------------------------------------------------------------------------
// Kernel 2: persistent sequential scan.  32 WGs x 128 threads (4 waves).
// WG g owns output columns [16g, 16g+16); wave w owns rows [16w, 16w+16).
// W_hh slice (16 cols x 512 K, padded) lives in LDS for the whole kernel.
// h is exchanged through a double-buffered global array with one atomic
// grid barrier per timestep.  4 accumulators break the K=512 WMMA chain
// into 4 independent 32-deep chains.
// ---------------------------------------------------------------------------
__global__ __launch_bounds__(128) void rnn_scan(const float* __restrict__ Whh,
                                                float* __restrict__ out,   // xp in, h out
                                                float* __restrict__ hbuf,  // [2][64][512]
                                                unsigned* __restrict__ counter) {
    __shared__ float wsl[16 * WPAD];

    const int tid  = threadIdx.x;
    const int wave = tid >> 5;
    const int lane = tid & 31;
    const int lr   = lane & 15;
    const int hi   = lane >> 4;
    const int n0   = blockIdx.x * 16;   // this WG's column slab
    const int m0   = wave * 16;         // this wave's row tile

    // Stage W_hh[n0..n0+15][0..511] into padded LDS (coalesced b32 loads).
    for (int idx = tid; idx < 16 * H_DIM; idx += 128) {
        const int n = idx >> 9;
        const int k = idx & 511;
        wsl[n * WPAD + k] = Whh[(size_t)(n0 + n) * H_DIM + k];
    }
    __syncthreads();

    float* h0 = hbuf;                       // zeroed by init_ws
    float* h1 = hbuf + B_DIM * H_DIM;
    const float* bbase = wsl + lr * WPAD + 2 * hi;   // B frag base (LDS)
    unsigned phase = 0;

    for (int t = 0; t < S_LEN; ++t) {
        const float* src = (t & 1) ? h1 : h0;
        float*       dst = (t & 1) ? h0 : h1;
        float*       xpt = out + (size_t)t * (B_DIM * H_DIM);

        // Seed accumulator 0 with the precomputed input projection tile.
        v8f c0, c1, c2, c3;
#pragma unroll
        for (int r = 0; r < 8; ++r)
            c0[r] = xpt[(size_t)(m0 + r + 8 * hi) * H_DIM + n0 + lr];
        c1 = (v8f){0,0,0,0,0,0,0,0};
        c2 = (v8f){0,0,0,0,0,0,0,0};
        c3 = (v8f){0,0,0,0,0,0,0,0};

        // Pull next timestep's xp tile toward this WGP while we compute.
        if (t + 1 < S_LEN) {
            const float* nxt = xpt + B_DIM * H_DIM + (size_t)(m0 + lr) * H_DIM + n0;
            __builtin_prefetch(nxt, 0, 1);
        }

        const float* arow = src + (size_t)(m0 + lr) * H_DIM + 2 * hi;
        for (int k0 = 0; k0 < H_DIM; k0 += 16) {
            v2f a0 = *(const v2f*)(arow + k0);
            v2f b0 = *(const v2f*)(bbase + k0);
            c0 = wmma_f32_k4(a0, b0, c0);
            v2f a1 = *(const v2f*)(arow + k0 + 4);
            v2f b1 = *(const v2f*)(bbase + k0 + 4);
            c1 = wmma_f32_k4(a1, b1, c1);
            v2f a2 = *(const v2f*)(arow + k0 + 8);
            v2f b2 = *(const v2f*)(bbase + k0 + 8);
            c2 = wmma_f32_k4(a2, b2, c2);
            v2f a3 = *(const v2f*)(arow + k0 + 12);
            v2f b3 = *(const v2f*)(bbase + k0 + 12);
            c3 = wmma_f32_k4(a3, b3, c3);
        }

#pragma unroll
        for (int r = 0; r < 8; ++r) {
            const float v = tanhf(c0[r] + c1[r] + c2[r] + c3[r]);
            const size_t off = (size_t)(m0 + r + 8 * hi) * H_DIM + n0 + lr;
            dst[off] = v;     // next step's h
            xpt[off] = v;     // final output (overwrites consumed xp slice)
        }

        // --- grid barrier: release h writes, arrive, spin, acquire ---
        __threadfence();
        __syncthreads();
        if (tid == 0) {
            atomicAdd(counter, 1u);
            const unsigned target = (phase + 1) * NWG;
            while (__hip_atomic_load(counter, __ATOMIC_RELAXED,
                                     __HIP_MEMORY_SCOPE_AGENT) < target)
                __builtin_amdgcn_s_sleep(1);
        }
        ++phase;
        __syncthreads();
        __threadfence();
    }
}

// ---------------------------------------------------------------------------
extern "C" void kernel_launch(void* const* d_in, const int* in_sizes, int n_in,
                              void* d_out, int out_size, void* d_ws, size_t ws_size,
                              hipStream_t stream) {
    (void)in_sizes; (void)n_in; (void)out_size; (void)ws_size;
    const float* x   = (const float*)d_in[0];   // [S,B,I]
    const float* Wih = (const float*)d_in[1];   // [H,I]
    const float* Whh = (const float*)d_in[2];   // [H,H]
    const float* bih = (const float*)d_in[3];   // [H]
    const float* bhh = (const float*)d_in[4];   // [H]
    float* out = (float*)d_out;                 // [S*B, H]

    float*    hbuf    = (float*)d_ws;                                   // 2*64*512 f32
    unsigned* counter = (unsigned*)((char*)d_ws +
                        (size_t)2 * B_DIM * H_DIM * sizeof(float));     // barrier ctr

    init_ws<<<(B_DIM * H_DIM + 255) / 256, 256, 0, stream>>>(hbuf, counter);

    // 1024 x 4 blocks of 128x128 output tiles
    dim3 grid(S_LEN * B_DIM / 128, H_DIM / 128, 1);
    xp_gemm<<<grid, 256, 0, stream>>>(x, Wih, bih, bhh, out);

    rnn_scan<<<NWG, 128, 0, stream>>>(Whh, out, hbuf, counter);
}